// FNO_multimodal_3d_80083960201368
// MI455X (gfx1250) — compile-verified
//
#include <hip/hip_runtime.h>

#define ALPHA 0.05f
#define PL    5184          // 72*72
#define VOL   373248        // 72^3
#define NCH   32
#define NV    (NCH*VOL)     // 11,943,936 floats
#define NFF   (NCH*72*72*8*2) // 2,654,208 floats (complex mode cube, kz<8)
#define NWT   (8*27648)     // transposed conv weights: 8 tensors x 27x32x32

typedef __attribute__((ext_vector_type(2))) float v2f;
typedef __attribute__((ext_vector_type(8))) float v8f;

__device__ __forceinline__ float leaky_f(float v) { return v > 0.f ? v : ALPHA * v; }

// CDNA5 async global->LDS copies (ASYNCcnt-tracked, no VGPR return)
__device__ __forceinline__ uint32_t lds_off32(const void* p) {
  return (uint32_t)(uintptr_t)p;    // low 32 bits of generic ptr = wave-relative LDS offset
}
__device__ __forceinline__ void async_b32(uint32_t ldsoff, const float* g) {
  asm volatile("global_load_async_to_lds_b32 %0, %1, off" :: "v"(ldsoff), "v"(g) : "memory");
}
__device__ __forceinline__ void async_b128(uint32_t ldsoff, const float* g) {
  asm volatile("global_load_async_to_lds_b128 %0, %1, off" :: "v"(ldsoff), "v"(g) : "memory");
}
__device__ __forceinline__ void async_wait0() {
  asm volatile("s_wait_asynccnt 0" ::: "memory");
}

// ---------------------------------------------------------------------------
// fc0: build 6 features, 6->32 linear, write padded NCDHW (zeros in pad band)
// ---------------------------------------------------------------------------
__global__ __launch_bounds__(256) void k_fc0(const float* __restrict__ ex,
                                             const float* __restrict__ ey,
                                             const float* __restrict__ ez,
                                             const float* __restrict__ w,
                                             const float* __restrict__ b,
                                             float* __restrict__ X) {
  int p = blockIdx.x * 256 + threadIdx.x;          // < VOL
  int z = p % 72, y = (p / 72) % 72, x = p / PL;
  if (x < 64 && y < 64 && z < 64) {
    int q = (x * 64 + y) * 64 + z;
    float f[6];
    f[0] = ex[q]; f[1] = ey[q]; f[2] = ez[q];
    f[3] = x * (1.f / 63.f); f[4] = y * (1.f / 63.f); f[5] = z * (1.f / 63.f);
    for (int c = 0; c < 32; ++c) {
      float v = b[c];
#pragma unroll
      for (int k = 0; k < 6; ++k) v += f[k] * w[k * 32 + c];
      X[c * VOL + p] = v;
    }
  } else {
    for (int c = 0; c < 32; ++c) X[c * VOL + p] = 0.f;
  }
}

// ---------------------------------------------------------------------------
// Transpose conv weights w[co][ci][tap] -> wt[tap][co][ci] (per tensor).
// Tensor index = layer*2 + conv (conv0 = blk_w1, conv1 = blk_w2).
// Makes each lane's WMMA B K-pair contiguous + 8B aligned.
// ---------------------------------------------------------------------------
__global__ __launch_bounds__(256) void k_wtrans(const float* __restrict__ w1,
                                               const float* __restrict__ w2,
                                               float* __restrict__ wt) {
  int e = blockIdx.x * 256 + threadIdx.x;          // < NWT
  int tensor = e / 27648;
  int r = e % 27648;
  int t = r / 1024;
  int co = (r >> 5) & 31;
  int ci = r & 31;
  const float* src = (tensor & 1) ? w2 : w1;
  src += (tensor >> 1) * 27648;
  wt[e] = src[(co * 32 + ci) * 27 + t];
}

// ---------------------------------------------------------------------------
// Forward real Z-DFT: X[c][x][y][z] -> F[c][x][y][kz<8] complex
// ---------------------------------------------------------------------------
__global__ __launch_bounds__(64) void k_dftz_fwd(const float* __restrict__ in,
                                                 float* __restrict__ out) {
  __shared__ __align__(16) float ls[576];
  const float* gsrc = in + blockIdx.x * 576;
  uint32_t l0 = lds_off32(&ls[0]);
  for (int i = threadIdx.x; i < 144; i += 64)      // 144 x b128 = 576 floats
    async_b128(l0 + i * 16, gsrc + i * 4);
  async_wait0();
  __syncthreads();
  int l = threadIdx.x >> 3, kz = threadIdx.x & 7;
  const float* line = &ls[l * 72];
  float ar = 0.f, ai = 0.f;
  float w0 = -6.28318530717958647692f * (float)kz / 72.f;
  for (int z = 0; z < 72; ++z) {
    float s, c; __sincosf(w0 * (float)z, &s, &c);
    ar += line[z] * c; ai += line[z] * s;
  }
  int L = blockIdx.x * 8 + l;
  out[(L * 8 + kz) * 2 + 0] = ar;
  out[(L * 8 + kz) * 2 + 1] = ai;
}

// ---------------------------------------------------------------------------
// Complex DFT along Y (axis 2 of [c][x][y][kz]); sign=-1 fwd, +1 inv
// ---------------------------------------------------------------------------
__global__ __launch_bounds__(72) void k_dfty(const float* __restrict__ in,
                                             float* __restrict__ out,
                                             float sign, float scale) {
  __shared__ __align__(16) float ls[1152];
  int base = blockIdx.x * 1152;                    // blockIdx = c*72 + x
  const float* gsrc = in + base;
  uint32_t l0 = lds_off32(&ls[0]);
#pragma unroll
  for (int j = 0; j < 4; ++j) {                    // 72 thr * 4 * 4 floats = 1152
    int e = threadIdx.x * 16 + j * 4;
    async_b128(l0 + e * 4, gsrc + e);
  }
  async_wait0();
  __syncthreads();
  int ky = threadIdx.x;
  float w0 = sign * 6.28318530717958647692f * (float)ky / 72.f;
  float aR[8], aI[8];
#pragma unroll
  for (int k = 0; k < 8; ++k) { aR[k] = 0.f; aI[k] = 0.f; }
  for (int y = 0; y < 72; ++y) {
    float s, c; __sincosf(w0 * (float)y, &s, &c);
    const float* py = &ls[y * 16];
#pragma unroll
    for (int k = 0; k < 8; ++k) {
      float re = py[2 * k], im = py[2 * k + 1];
      aR[k] += re * c - im * s;
      aI[k] += re * s + im * c;
    }
  }
  float* po = &out[base + ky * 16];
#pragma unroll
  for (int k = 0; k < 8; ++k) { po[2 * k] = aR[k] * scale; po[2 * k + 1] = aI[k] * scale; }
}

// ---------------------------------------------------------------------------
// Complex DFT along X (axis 1 of [c][x][y][kz]); sign=-1 fwd, +1 inv
// ---------------------------------------------------------------------------
__global__ __launch_bounds__(72) void k_dftx(const float* __restrict__ in,
                                             float* __restrict__ out,
                                             float sign, float scale) {
  __shared__ __align__(16) float ls[1152];
  int c = blockIdx.x / 72, ky = blockIdx.x % 72;
  const float* gbase = in + ((c * 72) * 72 + ky) * 16;   // row x: stride 1152 floats
  uint32_t l0 = lds_off32(&ls[0]);
#pragma unroll
  for (int j = 0; j < 4; ++j) {
    int f = j * 4;
    async_b128(l0 + (threadIdx.x * 16 + f) * 4, gbase + threadIdx.x * 1152 + f);
  }
  async_wait0();
  __syncthreads();
  int kx = threadIdx.x;
  float w0 = sign * 6.28318530717958647692f * (float)kx / 72.f;
  float aR[8], aI[8];
#pragma unroll
  for (int k = 0; k < 8; ++k) { aR[k] = 0.f; aI[k] = 0.f; }
  for (int x = 0; x < 72; ++x) {
    float s, c2; __sincosf(w0 * (float)x, &s, &c2);
    const float* px = &ls[x * 16];
#pragma unroll
    for (int k = 0; k < 8; ++k) {
      float re = px[2 * k], im = px[2 * k + 1];
      aR[k] += re * c2 - im * s;
      aI[k] += re * s + im * c2;
    }
  }
  float* po = &out[((c * 72 + kx) * 72 + ky) << 4];
#pragma unroll
  for (int k = 0; k < 8; ++k) { po[2 * k] = aR[k] * scale; po[2 * k + 1] = aI[k] * scale; }
}

// ---------------------------------------------------------------------------
// Inverse real Z transform (modes >=8 are zero, Hermitian half-spectrum)
// ---------------------------------------------------------------------------
__global__ __launch_bounds__(256) void k_dftz_inv(const float* __restrict__ in,
                                                  float* __restrict__ out) {
  int idx = blockIdx.x * 256 + threadIdx.x;        // < NV
  int z = idx % 72;
  int L = idx / 72;
  const float* X = &in[L * 16];
  float acc = X[0];
#pragma unroll
  for (int k = 1; k < 8; ++k) {
    float s, c;
    __sincosf(6.28318530717958647692f * (float)(k * z) / 72.f, &s, &c);
    acc += 2.f * (X[2 * k] * c - X[2 * k + 1] * s);
  }
  out[idx] = acc * (1.f / 72.f);
}

// ---------------------------------------------------------------------------
// Spectral mode-space 5x5x5 conv per octant block (block-local zero pad).
// ---------------------------------------------------------------------------
__global__ __launch_bounds__(256) void k_specconv(const float* __restrict__ F,
                                                  float* __restrict__ G,
                                                  const float* __restrict__ W) {
  __shared__ __align__(16) float lin[12800];
  const int X0[8]  = {0, 64, 0, 64, 8, 0, 8, 64};
  const int NXo[8] = {8, 8, 8, 8, 56, 8, 56, 8};
  const int Y0[8]  = {0, 0, 64, 64, 0, 8, 64, 8};
  const int NYo[8] = {8, 8, 8, 8, 8, 56, 8, 56};
  const int CUM[8] = {0, 64, 128, 192, 256, 704, 1152, 1600};
  int b = blockIdx.x, o = 0;
#pragma unroll
  for (int i = 1; i < 8; ++i) if (b >= CUM[i]) o = i;
  int li = b - CUM[o];
  int nx = NXo[o], ny = NYo[o], x0 = X0[o], y0 = Y0[o];
  int bx = li / ny, by = li % ny;

  uint32_t l0 = lds_off32(&lin[0]);
  for (int e = threadIdx.x; e < 12800; e += 256) {
    int kz = e & 7;
    int ic = (e >> 3) & 63;
    int dy = (e >> 9) % 5;
    int dx = e / 2560;
    int lx = bx + dx - 2, ly = by + dy - 2;
    if ((unsigned)lx < (unsigned)nx && (unsigned)ly < (unsigned)ny) {
      int ch = ic & 31, part = ic >> 5;
      async_b32(l0 + e * 4,
                &F[(((ch * 72 + (x0 + lx)) * 72 + (y0 + ly)) * 8 + kz) * 2 + part]);
    } else {
      lin[e] = 0.f;
    }
  }
  async_wait0();
  __syncthreads();

  int oc = threadIdx.x >> 2;
  int sub = threadIdx.x & 3;
  const float* wo = &W[(o * 64 + oc) * 8000];      // [ic][5][5][5]
#pragma unroll
  for (int j = 0; j < 2; ++j) {
    int bz = sub * 2 + j;
    float acc = 0.f;
    for (int dxy = 0; dxy < 25; ++dxy) {
      const float* lbase = &lin[dxy * 512];
      for (int ic = 0; ic < 64; ++ic) {
        const float* lv = &lbase[ic * 8];
        const float* wv = &wo[ic * 125 + dxy * 5];
#pragma unroll
        for (int dz = 0; dz < 5; ++dz) {
          int kz = bz + dz - 2;
          if ((unsigned)kz < 8u) acc += wv[dz] * lv[kz];
        }
      }
    }
    int cc = oc >> 1, part = oc & 1;
    G[(((cc * 72 + (x0 + bx)) * 72 + (y0 + by)) * 8 + bz) * 2 + part] = acc;
  }
}

// ---------------------------------------------------------------------------
// 3x3x3 conv, 32->32 ch, implicit GEMM with V_WMMA_F32_16X16X4_F32.
// LDS layout interleaves channel pairs so the A operand is one aligned
// ds_load_b64 (conflict-free: lane i hits banks 2i, 2i+1); weights are
// pre-transposed to wt[tap][co][ci] so B is one aligned global_load_b64.
// ---------------------------------------------------------------------------
__global__ __launch_bounds__(64) void k_conv3(const float* __restrict__ in,
                                              const float* __restrict__ wt,
                                              float* __restrict__ out,
                                              const float* __restrict__ add1,
                                              const float* __restrict__ addres,
                                              int leakyOut, int fuse) {
  __shared__ __align__(16) float lds[9 * 16 * 18 * 2]; // [col][ci/2][z(18)][2]
  const int px = blockIdx.x, py = blockIdx.y, zb = blockIdx.z * 16;
  const int tid = threadIdx.x;

  const int lane = tid & 31;
  const int n0 = (tid >> 5) * 16;                  // wave -> co tile
  const int hi = (lane >= 16) ? 1 : 0;             // lane half -> K pair
  const int m = lane & 15;                         // A row (z offset) / B col
  const int n = n0 + m;                            // this lane's co for B/store

  __builtin_prefetch(&wt[n * 32], 0, 0);           // warm first tap's weight row

  uint32_t l0 = lds_off32(&lds[0]);
  for (int e = tid; e < 9 * 32 * 18; e += 64) {
    int lz = e % 18;
    int ci = (e / 18) % 32;
    int col = e / (18 * 32);
    // pair-interleaved LDS slot for (col, ci, lz)
    int slot = ((col * 16 + (ci >> 1)) * 18 + lz) * 2 + (ci & 1);
    int gx = px + col / 3 - 1, gy = py + col % 3 - 1, gz = zb + lz - 1;
    if ((unsigned)gx < 72u && (unsigned)gy < 72u && (unsigned)gz < 72u)
      async_b32(l0 + slot * 4, &in[ci * VOL + gx * PL + gy * 72 + gz]);
    else
      lds[slot] = 0.f;
  }
  async_wait0();
  __syncthreads();

  v8f acc = {0.f, 0.f, 0.f, 0.f, 0.f, 0.f, 0.f, 0.f};

  for (int t = 0; t < 27; ++t) {
    const int col = t / 3, dz = t % 3;
    const float* lcol = &lds[col * (16 * 18 * 2)];
    const float* wrow = &wt[t * 1024 + n * 32];    // wt[t][n][ci], pair contiguous
#pragma unroll
    for (int kc = 0; kc < 8; ++kc) {
      const int cp = kc * 2 + hi;                  // channel-pair index
      v2f a = *(const v2f*)&lcol[(cp * 18 + m + dz) * 2];
      v2f bmat = *(const v2f*)&wrow[cp * 2];
      acc = __builtin_amdgcn_wmma_f32_16x16x4_f32(false, a, false, bmat,
                                                  (short)0, acc, false, false);
    }
  }

#pragma unroll
  for (int r = 0; r < 8; ++r) {
    int mm = r + hi * 8;                           // C/D layout: VGPR r -> M=r / r+8
    int z = zb + mm;
    if (z < 72) {
      int idx = n * VOL + px * PL + py * 72 + z;
      float v = acc[r];
      if (fuse) v += add1[idx] + addres[idx];
      if (leakyOut) v = leaky_f(v);
      out[idx] = v;
    }
  }
}

// ---------------------------------------------------------------------------
// Head: crop to 64^3, fc1(32->128)+leaky, fc2(128->6); weights async-staged
// ---------------------------------------------------------------------------
__global__ __launch_bounds__(256) void k_head(const float* __restrict__ X,
                                              const float* __restrict__ w1,
                                              const float* __restrict__ b1,
                                              const float* __restrict__ w2,
                                              const float* __restrict__ b2,
                                              float* __restrict__ out) {
  __shared__ __align__(16) float lw1[4096];
  __shared__ __align__(16) float lb1[128];
  __shared__ __align__(16) float lw2[768];
  __shared__ __align__(16) float lb2[8];
  {
    uint32_t o1 = lds_off32(&lw1[0]);
#pragma unroll
    for (int j = 0; j < 4; ++j) {                  // 256 thr * 4 * 4 = 4096
      int e = threadIdx.x * 16 + j * 4;
      async_b128(o1 + e * 4, w1 + e);
    }
    if (threadIdx.x < 192) async_b128(lds_off32(&lw2[0]) + threadIdx.x * 16, w2 + threadIdx.x * 4);
    if (threadIdx.x < 32)  async_b128(lds_off32(&lb1[0]) + threadIdx.x * 16, b1 + threadIdx.x * 4);
    if (threadIdx.x < 6)   lb2[threadIdx.x] = b2[threadIdx.x];
  }
  async_wait0();
  __syncthreads();

  int p = blockIdx.x * 256 + threadIdx.x;          // < 262144
  int z = p & 63, y = (p >> 6) & 63, x = p >> 12;
  float xv[32];
#pragma unroll
  for (int c = 0; c < 32; ++c) xv[c] = X[c * VOL + x * PL + y * 72 + z];
  float o[6];
#pragma unroll
  for (int t = 0; t < 6; ++t) o[t] = lb2[t];
  for (int j = 0; j < 128; ++j) {
    float h = lb1[j];
#pragma unroll
    for (int c = 0; c < 32; ++c) h += xv[c] * lw1[c * 128 + j];
    h = leaky_f(h);
#pragma unroll
    for (int t = 0; t < 6; ++t) o[t] += h * lw2[j * 6 + t];
  }
  float* po = &out[p * 6];
#pragma unroll
  for (int t = 0; t < 6; ++t) po[t] = o[t];
}

// ---------------------------------------------------------------------------
extern "C" void kernel_launch(void* const* d_in, const int* in_sizes, int n_in,
                              void* d_out, int out_size, void* d_ws, size_t ws_size,
                              hipStream_t stream) {
  (void)in_sizes; (void)n_in; (void)out_size; (void)ws_size;
  const float* yeex  = (const float*)d_in[0];
  const float* yeey  = (const float*)d_in[1];
  const float* yeez  = (const float*)d_in[2];
  const float* fc0_w = (const float*)d_in[3];
  const float* fc0_b = (const float*)d_in[4];
  const float* spec_w = (const float*)d_in[5];
  const float* blk_w1 = (const float*)d_in[6];
  const float* blk_w2 = (const float*)d_in[7];
  const float* fc1_w = (const float*)d_in[8];
  const float* fc1_b = (const float*)d_in[9];
  const float* fc2_w = (const float*)d_in[10];
  const float* fc2_b = (const float*)d_in[11];

  float* X  = (float*)d_ws;       // activation, 32 x 72^3
  float* A  = X + NV;             // spectral branch output x1
  float* Bt = A + NV;             // basic-block intermediate
  float* F1 = Bt + NV;            // mode buffer 1
  float* F2 = F1 + NFF;           // mode buffer 2
  float* WT = F2 + NFF;           // transposed conv weights, 8 x 27x32x32

  k_wtrans<<<NWT / 256, 256, 0, stream>>>(blk_w1, blk_w2, WT);
  k_fc0<<<1458, 256, 0, stream>>>(yeex, yeey, yeez, fc0_w, fc0_b, X);

  for (int i = 0; i < 4; ++i) {
    const float* sw = spec_w + (size_t)i * 8 * 64 * 64 * 125;
    const float* wt1 = WT + (size_t)(i * 2 + 0) * 27648;
    const float* wt2 = WT + (size_t)(i * 2 + 1) * 27648;

    // spectral branch: fwd DFTs -> mode conv -> inverse DFTs
    k_dftz_fwd<<<20736, 64, 0, stream>>>(X, F1);
    k_dfty<<<2304, 72, 0, stream>>>(F1, F2, -1.f, 1.f);
    k_dftx<<<2304, 72, 0, stream>>>(F2, F1, -1.f, 1.f);
    hipMemsetAsync(F2, 0, (size_t)NFF * sizeof(float), stream);
    k_specconv<<<2048, 256, 0, stream>>>(F1, F2, sw);
    k_dftx<<<2304, 72, 0, stream>>>(F2, F1, 1.f, 1.f / 72.f);
    k_dfty<<<2304, 72, 0, stream>>>(F1, F2, 1.f, 1.f / 72.f);
    k_dftz_inv<<<46656, 256, 0, stream>>>(F2, A);

    // local branch (WMMA implicit GEMM) + fused residual combine
    k_conv3<<<dim3(72, 72, 5), 64, 0, stream>>>(X, wt1, Bt, nullptr, nullptr, 1, 0);
    k_conv3<<<dim3(72, 72, 5), 64, 0, stream>>>(Bt, wt2, X, A, X, (i < 3) ? 1 : 0, 1);
  }

  k_head<<<1024, 256, 0, stream>>>(X, fc1_w, fc1_b, fc2_w, fc2_b, (float*)d_out);
}